// ContrastiveLossOptimizedV1_78451872629248
// MI455X (gfx1250) — compile-verified
//
#include <hip/hip_runtime.h>

typedef __attribute__((ext_vector_type(2))) float v2f;
typedef __attribute__((ext_vector_type(8))) float v8f;

#define THREADS 1024
#define WAVE 32
#define NQ 5   // quantities: sum p, sum p^2, sum p*mask, sum p^2*mask, sum mask

__global__ __launch_bounds__(THREADS)
void contrastive_loss_wmma_kernel(const float* __restrict__ y_pred,
                                  const int*   __restrict__ y_true,
                                  float* __restrict__ out,
                                  int n)
{
    __shared__ float part[NQ][THREADS];

    const int tid  = threadIdx.x;
    const int lane = tid & (WAVE - 1);

    // ---- Stage A: per-thread register accumulation (memory-bound part) ----
    float s1 = 0.f, s2 = 0.f, sp = 0.f, sq = 0.f, c = 0.f;
    for (int i = tid; i < n; i += THREADS) {
        float x  = y_pred[i];
        float p  = 1.0f / (1.0f + expf(-x));   // sigmoid
        float p2 = p * p;
        float m  = (y_true[i] != 0) ? 1.0f : 0.0f;
        s1 += p;
        s2 += p2;
        sp += m * p;
        sq += m * p2;
        c  += m;
    }
    part[0][tid] = s1;
    part[1][tid] = s2;
    part[2][tid] = sp;
    part[3][tid] = sq;
    part[4][tid] = c;
    __syncthreads();

    // ---- Stage B: WMMA-based cross-thread reduction ----
    // A = ones(16x4); each v_wmma_f32_16x16x4_f32 reduces a 64-value chunk of
    // partials (B = 4x16) into the f32 accumulator D. D ends up holding the
    // grand total replicated over its 16 rows, so (sum of all D entries)/16
    // equals the total regardless of the exact VGPR lane layout of A/B.
    // All waves do this redundantly on identical LDS data -> EXEC all-ones.
    v2f a;
    a[0] = 1.0f;
    a[1] = 1.0f;

    float tot[NQ];
    #pragma unroll
    for (int q = 0; q < NQ; ++q) {
        v8f acc = {};   // zero C accumulator
        #pragma unroll
        for (int t = 0; t < THREADS / 64; ++t) {
            v2f b;
            b[0] = part[q][t * 64 + lane];
            b[1] = part[q][t * 64 + 32 + lane];
            acc = __builtin_amdgcn_wmma_f32_16x16x4_f32(
                /*neg_a=*/false, a,
                /*neg_b=*/false, b,
                /*c_mod=*/(short)0, acc,
                /*reuse_a=*/false, /*reuse_b=*/false);
        }
        float r = acc[0] + acc[1] + acc[2] + acc[3] +
                  acc[4] + acc[5] + acc[6] + acc[7];
        // sum over the 32 lanes of this wave
        #pragma unroll
        for (int off = 16; off >= 1; off >>= 1)
            r += __shfl_xor(r, off, WAVE);
        tot[q] = r * (1.0f / 16.0f);   // 16 identical rows in D
    }

    // ---- Stage C: closed-form final loss (thread 0) ----
    if (tid == 0) {
        const float S1 = tot[0];          // sum p
        const float S2 = tot[1];          // sum p^2
        const float SP = tot[2];          // sum p over positives
        const float SQ = tot[3];          // sum p^2 over positives
        const float NP = tot[4];          // count of positives
        const float Nf = (float)n;
        const float NN = Nf - NP;         // count of negatives

        // sum_{i,j} (p_j - p_i)^2 = 2*N*S2 - 2*S1^2
        const float sum_dist_sq = 2.0f * Nf * S2 - 2.0f * S1 * S1;

        const float sum_neg    = S1 - SP;
        const float sum_neg_sq = S2 - SQ;

        float ss_pos = 0.0f;
        if (NP > 0.0f) ss_pos = SQ - (SP * SP) / NP;
        float ss_neg = 0.0f;
        if (NN > 0.0f) ss_neg = sum_neg_sq - (sum_neg * sum_neg) / NN;

        // mean(indicator) = 2 * n_pos * n_neg / N^2
        const float f = 2.0f * NP * NN / (Nf * Nf);

        out[0] = sum_dist_sq * f + (ss_pos + ss_neg) * (1.0f - f);
    }
}

extern "C" void kernel_launch(void* const* d_in, const int* in_sizes, int n_in,
                              void* d_out, int out_size, void* d_ws, size_t ws_size,
                              hipStream_t stream) {
    const float* y_pred = (const float*)d_in[0];
    const int*   y_true = (const int*)d_in[1];
    // d_in[2] = epoch: unused by the reference math
    float* out = (float*)d_out;
    const int n = in_sizes[0];

    contrastive_loss_wmma_kernel<<<1, THREADS, 0, stream>>>(y_pred, y_true, out, n);
}